// TopoLoss_24026047054261
// MI455X (gfx1250) — compile-verified
//
#include <hip/hip_runtime.h>
#include <hip/hip_bf16.h>
#include <math.h>

typedef float v2f __attribute__((ext_vector_type(2)));
typedef float v8f __attribute__((ext_vector_type(8)));

#define MAXN 2048

// ---------------------------------------------------------------------------
// Row squared-norm: sqn[row] = sum_k X[row,k]^2.  grid = n blocks, 256 thr.
// ---------------------------------------------------------------------------
__global__ __launch_bounds__(256)
void sqnorm_kernel(const float* __restrict__ X, float* __restrict__ sqn, int K) {
  const int row = blockIdx.x;
  const float4* p = (const float4*)(X + (size_t)row * K);
  const int K4 = K >> 2;
  float s = 0.f;
  for (int k = threadIdx.x; k < K4; k += 256) {
    float4 v = p[k];
    s += v.x * v.x + v.y * v.y + v.z * v.z + v.w * v.w;
  }
  for (int off = 16; off; off >>= 1) s += __shfl_down(s, off, 32);
  __shared__ float ws[8];
  if ((threadIdx.x & 31) == 0) ws[threadIdx.x >> 5] = s;
  __syncthreads();
  if (threadIdx.x == 0) {
    float t = 0.f;
    for (int i = 0; i < 8; ++i) t += ws[i];
    sqn[row] = t;
  }
}

// ---------------------------------------------------------------------------
// Pairwise distances via gram trick, fp32 WMMA 16x16x4.
// Block = 128 threads (4 waves); block tile 128x128; wave tile 64x64.
// Symmetric: only upper-triangular blocks (bx >= by) compute; each fragment
// is written twice: direct (row-coalesced b32 across lanes) and mirrored
// (8 consecutive M values per lane -> two b128 stores per lane).
// D[i,j] = sqrt(max(sq[i] + sq[j] - 2*(X X^T)[i,j], 0)); also atomicMax of D.
// ---------------------------------------------------------------------------
__global__ __launch_bounds__(128)
void dist_kernel(const float* __restrict__ X, const float* __restrict__ sqn,
                 float* __restrict__ D, unsigned* __restrict__ maxbits,
                 int n, int K) {
  if (blockIdx.x < blockIdx.y) return;  // lower-triangular blocks: nothing to do

  const int lane = threadIdx.x & 31;
  const int wave = threadIdx.x >> 5;
  const int r0 = blockIdx.y * 128 + (wave >> 1) * 64;
  const int c0 = blockIdx.x * 128 + (wave & 1) * 64;
  const int lrow  = lane & 15;         // row-in-16 (A) / col-in-16 (B)
  const int koff2 = (lane >> 4) << 1;  // lanes 0-15 -> K=0,1 ; lanes 16-31 -> K=2,3

  v8f acc[4][4];
  const v8f vzero = {};
#pragma unroll
  for (int i = 0; i < 4; ++i)
#pragma unroll
    for (int j = 0; j < 4; ++j) acc[i][j] = vzero;

  for (int kk = 0; kk < K; kk += 4) {
    v2f a[4], b[4];
    const int kb = kk + koff2;
#pragma unroll
    for (int g = 0; g < 4; ++g)
      a[g] = *reinterpret_cast<const v2f*>(X + (size_t)(r0 + g * 16 + lrow) * K + kb);
#pragma unroll
    for (int g = 0; g < 4; ++g)
      b[g] = *reinterpret_cast<const v2f*>(X + (size_t)(c0 + g * 16 + lrow) * K + kb);
#pragma unroll
    for (int i = 0; i < 4; ++i)
#pragma unroll
      for (int j = 0; j < 4; ++j)
        acc[i][j] = __builtin_amdgcn_wmma_f32_16x16x4_f32(
            false, a[i], false, b[j], (short)0, acc[i][j], false, false);
  }

  // Epilogue. C/D layout: M = v + 8*(lane>>4), N = lane&15 (VGPR v of 8).
  const int mbase = (lane >> 4) * 8;
  const int ncol  = lane & 15;
  float lmax = 0.f;
#pragma unroll
  for (int ti = 0; ti < 4; ++ti) {
    const int gi0 = r0 + ti * 16 + mbase;
    float si[8];
#pragma unroll
    for (int v = 0; v < 8; ++v) si[v] = sqn[gi0 + v];
#pragma unroll
    for (int tj = 0; tj < 4; ++tj) {
      const int gj = c0 + tj * 16 + ncol;
      const float sj = sqn[gj];
      float dd8[8];
#pragma unroll
      for (int v = 0; v < 8; ++v) {
        float d2 = si[v] + sj - 2.f * acc[ti][tj][v];
        float dd = sqrtf(fmaxf(d2, 0.f));
        dd8[v] = dd;
        D[(size_t)(gi0 + v) * n + gj] = dd;  // direct (upper) half
        lmax = fmaxf(lmax, dd);
      }
      // mirrored (lower) half: 8 consecutive columns -> two b128 stores
      float4* mp = (float4*)(D + (size_t)gj * n + gi0);
      mp[0] = make_float4(dd8[0], dd8[1], dd8[2], dd8[3]);
      mp[1] = make_float4(dd8[4], dd8[5], dd8[6], dd8[7]);
    }
  }
  // block max -> global atomic (float bits monotonic for non-negative floats)
  __shared__ unsigned smax;
  if (threadIdx.x == 0) smax = 0u;
  __syncthreads();
  atomicMax(&smax, __float_as_uint(lmax));
  __syncthreads();
  if (threadIdx.x == 0) atomicMax(maxbits, smax);
}

// ---------------------------------------------------------------------------
// Prim's MST, one workgroup of 1024 threads (32 waves). Packed u64 argmin
// (first-index tiebreak matches jnp.argmin). 2 barriers per iteration:
// after the per-wave leaders publish wmin[32], every wave redundantly
// reduces the 32 entries and derives j locally (no broadcast barrier).
// ---------------------------------------------------------------------------
__global__ __launch_bounds__(1024)
void mst_kernel(const float* __restrict__ D, int n, int2* __restrict__ edges) {
  __shared__ float md[MAXN];
  __shared__ int   par[MAXN];
  __shared__ unsigned long long wmin[32];

  const int t = threadIdx.x;
  const int lane = t & 31;
  const float INF = __uint_as_float(0x7f800000u);

  for (int i = t; i < n; i += 1024) {
    md[i]  = (i == 0) ? INF : D[i];  // row 0, with min_dist[0] = inf
    par[i] = 0;
  }
  __syncthreads();

  for (int it = 0; it < n - 1; ++it) {
    // ---- local + wave argmin over md ----
    unsigned long long best = ~0ull;
    for (int i = t; i < n; i += 1024) {
      unsigned long long p =
          ((unsigned long long)__float_as_uint(md[i]) << 32) | (unsigned)i;
      best = (p < best) ? p : best;
    }
    for (int off = 16; off; off >>= 1) {
      unsigned long long o = __shfl_down(best, off, 32);
      best = (o < best) ? o : best;
    }
    if (lane == 0) wmin[t >> 5] = best;
    __syncthreads();
    // ---- every wave reduces the 32 partials; all threads derive j ----
    unsigned long long b = wmin[lane];
    for (int off = 16; off; off >>= 1) {
      unsigned long long o = __shfl_down(b, off, 32);
      b = (o < b) ? o : b;
    }
    b = __shfl(b, 0, 32);
    const int j = (int)(b & 0xffffffffu);
    if (t == 0) edges[it] = make_int2(par[j], j);  // par[j] untouched by relax
    // ---- relax against row j ----
    const float* row = D + (size_t)j * n;
    for (int i = t; i < n; i += 1024) {
      if (i == j) {
        md[i] = INF;
      } else {
        float dj = row[i];
        if (dj < md[i]) { md[i] = dj; par[i] = j; }
      }
    }
    __syncthreads();
  }
}

// ---------------------------------------------------------------------------
// Gather MST edges from both matrices, compute data_loss + latent_loss.
// Normalization applied here (MST edges are scale-invariant, values are not).
// ---------------------------------------------------------------------------
__global__ __launch_bounds__(256)
void loss_kernel(const float* __restrict__ Dd, const float* __restrict__ Dl,
                 const int2* __restrict__ ed, const int2* __restrict__ el,
                 int nedge, int n, const unsigned* __restrict__ maxbits,
                 const float* __restrict__ latent_norm,
                 float* __restrict__ total) {
  const float inv_max = 1.f / __uint_as_float(maxbits[0]);
  const float inv_ln  = 1.f / latent_norm[0];
  float s0 = 0.f, s1 = 0.f;
  for (int e = threadIdx.x; e < nedge; e += 256) {
    int2 E = ed[e];
    size_t o = (size_t)E.x * n + E.y;
    float df = Dd[o] * inv_max - Dl[o] * inv_ln;
    s0 += df * df;
    E = el[e];
    o = (size_t)E.x * n + E.y;
    df = Dd[o] * inv_max - Dl[o] * inv_ln;
    s1 += df * df;
  }
  for (int off = 16; off; off >>= 1) {
    s0 += __shfl_down(s0, off, 32);
    s1 += __shfl_down(s1, off, 32);
  }
  __shared__ float w0[8], w1[8];
  if ((threadIdx.x & 31) == 0) { w0[threadIdx.x >> 5] = s0; w1[threadIdx.x >> 5] = s1; }
  __syncthreads();
  if (threadIdx.x == 0) {
    float a = 0.f, b = 0.f;
    for (int i = 0; i < 8; ++i) { a += w0[i]; b += w1[i]; }
    total[0] = a / (float)nedge + b / (float)nedge;  // data_loss + latent_loss
  }
}

// ---------------------------------------------------------------------------
// ae_loss[row] = mean_k (y_true - y_pred)^2.  grid = n blocks, 256 threads.
// ---------------------------------------------------------------------------
__global__ __launch_bounds__(256)
void rowmse_kernel(const float* __restrict__ A, const float* __restrict__ B,
                   float* __restrict__ out, int K) {
  const int row = blockIdx.x;
  const float4* pa = (const float4*)(A + (size_t)row * K);
  const float4* pb = (const float4*)(B + (size_t)row * K);
  const int K4 = K >> 2;
  float s = 0.f;
  for (int k = threadIdx.x; k < K4; k += 256) {
    float4 a = pa[k], b = pb[k];
    float dx = a.x - b.x, dy = a.y - b.y, dz = a.z - b.z, dw = a.w - b.w;
    s += dx * dx + dy * dy + dz * dz + dw * dw;
  }
  for (int off = 16; off; off >>= 1) s += __shfl_down(s, off, 32);
  __shared__ float ws[8];
  if ((threadIdx.x & 31) == 0) ws[threadIdx.x >> 5] = s;
  __syncthreads();
  if (threadIdx.x == 0) {
    float t = 0.f;
    for (int i = 0; i < 8; ++i) t += ws[i];
    out[row] = t / (float)K;
  }
}

__global__ void init_kernel(unsigned* maxbits) {
  maxbits[0] = 0u;
  maxbits[1] = 0u;
}

__global__ __launch_bounds__(256)
void finalize_kernel(const float* __restrict__ ae, const float* __restrict__ total,
                     float* __restrict__ out, int n) {
  int i = blockIdx.x * 256 + threadIdx.x;
  if (i < n) out[i] = ae[i] + 0.5f * total[0];  // REG_LAMBDA = 0.5
}

// ---------------------------------------------------------------------------
extern "C" void kernel_launch(void* const* d_in, const int* in_sizes, int n_in,
                              void* d_out, int out_size, void* d_ws, size_t ws_size,
                              hipStream_t stream) {
  const float* y_true      = (const float*)d_in[0];
  const float* latent      = (const float*)d_in[1];
  const float* y_pred      = (const float*)d_in[2];
  const float* latent_norm = (const float*)d_in[3];
  float* out = (float*)d_out;

  const int n  = out_size;          // 2048
  const int Kd = in_sizes[0] / n;   // 4096
  const int Kl = in_sizes[1] / n;   // 128

  char* ws = (char*)d_ws;
  size_t off = 0;
  auto alloc = [&](size_t bytes) -> void* {
    void* p = ws + off;
    off = (off + bytes + 255) & ~(size_t)255;
    return p;
  };
  float*    Dd    = (float*)alloc((size_t)n * n * sizeof(float));
  float*    Dl    = (float*)alloc((size_t)n * n * sizeof(float));
  float*    sqn_d = (float*)alloc((size_t)n * sizeof(float));
  float*    sqn_l = (float*)alloc((size_t)n * sizeof(float));
  float*    ae    = (float*)alloc((size_t)n * sizeof(float));
  int2*     ed    = (int2*)alloc((size_t)(n - 1) * sizeof(int2));
  int2*     el    = (int2*)alloc((size_t)(n - 1) * sizeof(int2));
  unsigned* maxb  = (unsigned*)alloc(2 * sizeof(unsigned));
  float*    total = (float*)alloc(sizeof(float));

  init_kernel<<<1, 1, 0, stream>>>(maxb);
  sqnorm_kernel<<<n, 256, 0, stream>>>(y_true, sqn_d, Kd);
  sqnorm_kernel<<<n, 256, 0, stream>>>(latent, sqn_l, Kl);

  dim3 grid(n / 128, n / 128);  // blocks with bx < by early-exit
  dist_kernel<<<grid, 128, 0, stream>>>(y_true, sqn_d, Dd, maxb + 0, n, Kd);
  dist_kernel<<<grid, 128, 0, stream>>>(latent, sqn_l, Dl, maxb + 1, n, Kl);

  mst_kernel<<<1, 1024, 0, stream>>>(Dd, n, ed);
  mst_kernel<<<1, 1024, 0, stream>>>(Dl, n, el);

  loss_kernel<<<1, 256, 0, stream>>>(Dd, Dl, ed, el, n - 1, n, maxb, latent_norm, total);
  rowmse_kernel<<<n, 256, 0, stream>>>(y_true, y_pred, ae, Kd);
  finalize_kernel<<<(n + 255) / 256, 256, 0, stream>>>(ae, total, out, n);
}